// MotionNet_30124900614668
// MI455X (gfx1250) — compile-verified
//
#include <hip/hip_runtime.h>

// ---------------------------------------------------------------------------
// MotionNet forward for MI455X (gfx1250, wave32).
// Bandwidth-bound (~60MB fp32 weights, ~2.2 GFLOP) -> single-pass fp32 reads,
// matrix math on V_WMMA_F32_16X16X4_F32 (16x16 C/D tiles, K-step 4).
// Ragged-N handled by clamping load columns (no exec-mask churn in the hot
// loop); invalid output columns are simply not stored.
// ---------------------------------------------------------------------------

typedef float v2f __attribute__((ext_vector_type(2)));
typedef float v8f __attribute__((ext_vector_type(8)));

__device__ __forceinline__ float elu_f(float x) {
    return x > 0.0f ? x : (__expf(x) - 1.0f);
}

// OUT(64 x N) = act( A(64 x K) @ W + bias ), W either (K,N) row-major
// (TRANS=false) or (N,K) row-major (TRANS=true, the 'ehi' einsum weights).
// Block = 128 threads = 4 waves; each wave computes one 16x16 tile.
// blockIdx.x: N/64 tile groups, blockIdx.y: M/16 (=4), blockIdx.z: expert.
template<bool TRANS, int ACT>
__global__ __launch_bounds__(128)
void gemm16(const float* __restrict__ A, int lda, int colA,
            const float* __restrict__ Wm, int ldw, long wStride,
            const float* __restrict__ bias,
            float* __restrict__ OUT, int ldo, int colO, long oStride,
            int N, int K)
{
    const int lane = threadIdx.x & 31;
    const int wave = threadIdx.x >> 5;
    const int half = lane >> 4;        // 0: lanes 0-15, 1: lanes 16-31
    const int lid  = lane & 15;
    const int koff = half * 2;         // K sub-offset held by this lane (A & B layouts)
    const int mBase = blockIdx.y * 16;
    const int nCol  = blockIdx.x * 64 + wave * 16 + lid;
    const float* W  = Wm  + (size_t)blockIdx.z * (size_t)wStride;
    float*       Ot = OUT + (size_t)blockIdx.z * (size_t)oStride;
    const bool nOK  = nCol < N;
    // Clamp: OOB lanes compute a duplicate of column N-1 (discarded at store).
    const int nLd   = nOK ? nCol : (N - 1);
    const int rowA  = mBase + lid;

    // Unconditional base pointers for the hot loop.
    const float* aPtr = A + (size_t)rowA * lda + colA + koff;
    const float* bPtr = TRANS ? (W + (size_t)nLd * ldw + koff)
                              : (W + (size_t)koff * ldw + nLd);

    v8f acc = {};
    #pragma unroll 4
    for (int k = 0; k < K; k += 4) {
        v2f a, b;
        // A fragment: 16x4 f32 -> 2 VGPRs/lane; contiguous pair -> b64 load.
        a = *(const v2f*)(aPtr);
        if (TRANS) {
            // Contiguous along K -> b64 load.
            b = *(const v2f*)(bPtr);
            bPtr += 4;
        } else {
            // Row-strided; two coalesced b32 loads.
            b.x = bPtr[0];
            b.y = bPtr[(size_t)ldw];
            bPtr += (size_t)4 * ldw;
        }
        aPtr += 4;
        acc = __builtin_amdgcn_wmma_f32_16x16x4_f32(
                  false, a, false, b, (short)0, acc, false, false);
    }

    if (nOK) {
        const float bs = bias ? bias[nCol] : 0.0f;
        #pragma unroll
        for (int r = 0; r < 8; ++r) {
            const int m = mBase + r + half * 8;   // C/D layout: VGPR r -> M=r / r+8
            float v = acc[r] + bs;
            if (ACT == 1) v = elu_f(v);
            Ot[(size_t)m * ldo + colO + nCol] = v;
        }
    }
}

// Softmax over 10 logits per batch row -> BC (64 x 10)
__global__ void softmax10(const float* __restrict__ logits, int ld,
                          float* __restrict__ BC)
{
    const int b = blockIdx.x * blockDim.x + threadIdx.x;
    if (b >= 64) return;
    float v[10];
    float mx = -3.4e38f;
    #pragma unroll
    for (int e = 0; e < 10; ++e) { v[e] = logits[b * ld + e]; mx = fmaxf(mx, v[e]); }
    float sum = 0.0f;
    #pragma unroll
    for (int e = 0; e < 10; ++e) { v[e] = __expf(v[e] - mx); sum += v[e]; }
    const float inv = 1.0f / sum;
    #pragma unroll
    for (int e = 0; e < 10; ++e) BC[b * 10 + e] = v[e] * inv;
}

// out[b,h] = act( sum_e BC[b,e] * (Y[b, e*estride + h] + eb[e,h]) )
template<int ACT>
__global__ void combine10(const float* __restrict__ Y, int ldy, int estride,
                          const float* __restrict__ eb, int ebld,
                          const float* __restrict__ BC,
                          float* __restrict__ out, int ldo, int Hdim)
{
    const int h = blockIdx.x * blockDim.x + threadIdx.x;
    const int b = blockIdx.y;
    if (h >= Hdim) return;
    float s = 0.0f;
    #pragma unroll
    for (int e = 0; e < 10; ++e)
        s += BC[b * 10 + e] * (Y[(size_t)b * ldy + e * estride + h] + eb[e * ebld + h]);
    if (ACT == 1) s = elu_f(s);
    out[(size_t)b * ldo + h] = s;
}

static void run_gemm(bool trans, bool act,
                     const float* A, int lda, int colA,
                     const float* W, int ldw, long wStride, const float* bias,
                     float* OUT, int ldo, int colO, long oStride,
                     int N, int K, int E, hipStream_t s)
{
    dim3 grid((unsigned)((N + 63) / 64), 4u, (unsigned)E);
    dim3 blk(128);
    if (trans) {
        if (act) gemm16<true, 1><<<grid, blk, 0, s>>>(A, lda, colA, W, ldw, wStride, bias, OUT, ldo, colO, oStride, N, K);
        else     gemm16<true, 0><<<grid, blk, 0, s>>>(A, lda, colA, W, ldw, wStride, bias, OUT, ldo, colO, oStride, N, K);
    } else {
        if (act) gemm16<false,1><<<grid, blk, 0, s>>>(A, lda, colA, W, ldw, wStride, bias, OUT, ldo, colO, oStride, N, K);
        else     gemm16<false,0><<<grid, blk, 0, s>>>(A, lda, colA, W, ldw, wStride, bias, OUT, ldo, colO, oStride, N, K);
    }
}

extern "C" void kernel_launch(void* const* d_in, const int* in_sizes, int n_in,
                              void* d_out, int out_size, void* d_ws, size_t ws_size,
                              hipStream_t stream)
{
    (void)in_sizes; (void)n_in; (void)out_size; (void)ws_size;

    // Model constants
    const int NIN = 4352, SGT = 4224, HG = 512, NE = 10, H = 512, ZD = 1792, NOUT = 647;
    const int starts[5] = {0, 512, 2560, 3584, 4096};
    const int din[5]    = {512, 2048, 1024, 512, 128};
    const int dh[5]     = {512, 128, 512, 512, 128};
    const int zoff[5]   = {0, 512, 640, 1152, 1664};

    const float* x   = (const float*)d_in[0];
    const float* gw0 = (const float*)d_in[1];
    const float* gb0 = (const float*)d_in[2];
    const float* gw1 = (const float*)d_in[3];
    const float* gb1 = (const float*)d_in[4];
    const float* gw2 = (const float*)d_in[5];
    const float* gb2 = (const float*)d_in[6];
    const float* ew0 = (const float*)d_in[27];
    const float* eb0 = (const float*)d_in[28];
    const float* ew1 = (const float*)d_in[29];
    const float* eb1 = (const float*)d_in[30];
    const float* ew2 = (const float*)d_in[31];
    const float* eb2 = (const float*)d_in[32];
    float* out = (float*)d_out;

    // Workspace layout (floats)
    float* ws     = (float*)d_ws;
    float* gH0    = ws;                  // 64*512
    float* gH1    = gH0    + 64 * 512;   // 64*512
    float* logits = gH1    + 64 * 512;   // 64*16 (padded)
    float* BC     = logits + 64 * 16;    // 64*10 (padded to 64*16)
    float* tmpE   = BC     + 64 * 16;    // 64*512
    float* zbuf   = tmpE   + 64 * 512;   // 64*1792
    float* h1     = zbuf   + 64 * ZD;    // 64*512
    float* h2     = h1     + 64 * 512;   // 64*512
    float* Y      = h2     + 64 * 512;   // 64*6560 (reused across mixture layers)

    // ---- Gating MLP ----
    run_gemm(false, true,  x,   NIN, SGT, gw0, HG, 0, gb0, gH0,    HG, 0, 0, HG, 128, 1, stream);
    run_gemm(false, true,  gH0, HG,  0,   gw1, HG, 0, gb1, gH1,    HG, 0, 0, HG, HG,  1, stream);
    run_gemm(false, false, gH1, HG,  0,   gw2, NE, 0, gb2, logits, 16, 0, 0, NE, HG,  1, stream);
    softmax10<<<dim3(1), dim3(64), 0, stream>>>(logits, 16, BC);

    // ---- Segment encoders -> z (64 x 1792) ----
    for (int i = 0; i < 5; ++i) {
        const float* w0 = (const float*)d_in[7 + 4 * i];
        const float* b0 = (const float*)d_in[8 + 4 * i];
        const float* w1 = (const float*)d_in[9 + 4 * i];
        const float* b1 = (const float*)d_in[10 + 4 * i];
        run_gemm(false, true, x,    NIN,   starts[i], w0, dh[i], 0, b0, tmpE, dh[i], 0,       0, dh[i], din[i], 1, stream);
        run_gemm(false, true, tmpE, dh[i], 0,         w1, dh[i], 0, b1, zbuf, ZD,    zoff[i], 0, dh[i], dh[i],  1, stream);
    }

    // ---- Mixture layer 1: per-expert z @ ew0[e]^T, then BC blend + eb0, ELU ----
    run_gemm(true, false, zbuf, ZD, 0, ew0, ZD, (long)H * ZD, nullptr,
             Y, NE * H, 0, H, H, ZD, NE, stream);
    combine10<1><<<dim3(2, 64), dim3(256), 0, stream>>>(Y, NE * H, H, eb0, H, BC, h1, H, H);

    // ---- Mixture layer 2 ----
    run_gemm(true, false, h1, H, 0, ew1, H, (long)H * H, nullptr,
             Y, NE * H, 0, H, H, H, NE, stream);
    combine10<1><<<dim3(2, 64), dim3(256), 0, stream>>>(Y, NE * H, H, eb1, H, BC, h2, H, H);

    // ---- Mixture output layer (N = 647, padded expert stride 656) ----
    run_gemm(true, false, h2, H, 0, ew2, H, (long)NOUT * H, nullptr,
             Y, NE * 656, 0, 656, NOUT, H, NE, stream);
    combine10<0><<<dim3(3, 64), dim3(256), 0, stream>>>(Y, NE * 656, 656, eb2, NOUT, BC, out, NOUT, NOUT);
}